// GaussianSplat_85529978732990
// MI455X (gfx1250) — compile-verified
//
#include <hip/hip_runtime.h>
#include <hip/hip_bf16.h>
#include <math.h>

// ---------------------------------------------------------------------------
// Gaussian splat renderer for gfx1250 (MI455X).
// sigma[p,g] = P[p,:] . G[g,:]  (rank-6 factorization, padded to K=8)
// computed per 16x16 (gaussian x pixel) tile with two V_WMMA_F32_16X16X4_F32.
// Per-block shared gaussian tiles are staged global->LDS with the CDNA5
// async DMA path (GLOBAL_LOAD_ASYNC_TO_LDS_*, ASYNCcnt), double buffered.
// ---------------------------------------------------------------------------

typedef __attribute__((ext_vector_type(2))) float v2f;
typedef __attribute__((ext_vector_type(8))) float v8f;

#define SH_C0      0.28209479177387814f
#define EPS2D      0.3f
#define NEAR_Z     0.01f
#define ALPHA_MIN  (1.0f / 255.0f)
#define ALPHA_MAX  0.999f
#define BIG_KEY    1.0e30f
#define SORT_CAP   2048

// ---------------------------------------------------------------------------
// Kernel 1: per-gaussian preprocessing -> packed WMMA "A" features, colors, key
// feat[n*8 + k]:
//   k0 = 0.5A u^2 + 0.5C v^2 + B u v - ln(opac)   (pixel feature 1)
//   k1 = -(A u + B v)                             (pixel feature px)
//   k2 = 0.5A                                     (pixel feature px^2)
//   k3 = -(C v + B u)                             (pixel feature py)
//   k4 = 0.5C                                     (pixel feature py^2)
//   k5 = B                                        (pixel feature px*py)
//   k6 = k7 = 0
// invalid gaussian -> k0 = BIG (alpha underflows to 0)
// ---------------------------------------------------------------------------
__global__ void gs_preprocess(const float* __restrict__ means,
                              const float* __restrict__ quats,
                              const float* __restrict__ scales,
                              const float* __restrict__ opacities,
                              const float* __restrict__ sh,
                              const float* __restrict__ viewmats,
                              const float* __restrict__ Ks,
                              int N,
                              float* __restrict__ feat,   // [N*8]
                              float* __restrict__ col4,   // [N*4]
                              float* __restrict__ key)    // [N]
{
    int n = blockIdx.x * blockDim.x + threadIdx.x;
    if (n >= N) return;

    // quaternion -> rotation
    float qw = quats[n * 4 + 0], qx = quats[n * 4 + 1];
    float qy = quats[n * 4 + 2], qz = quats[n * 4 + 3];
    float qn = rsqrtf(qw * qw + qx * qx + qy * qy + qz * qz);
    qw *= qn; qx *= qn; qy *= qn; qz *= qn;

    float sx = fminf(__expf(scales[n * 3 + 0]), 10.0f);
    float sy = fminf(__expf(scales[n * 3 + 1]), 10.0f);
    float sz = fminf(__expf(scales[n * 3 + 2]), 10.0f);
    float opac = 1.0f / (1.0f + __expf(-opacities[n]));

    float c0 = fmaxf(sh[n * 3 + 0] * SH_C0 + 0.5f, 0.0f);
    float c1 = fmaxf(sh[n * 3 + 1] * SH_C0 + 0.5f, 0.0f);
    float c2 = fmaxf(sh[n * 3 + 2] * SH_C0 + 0.5f, 0.0f);

    float R00 = 1.f - 2.f * (qy * qy + qz * qz);
    float R01 = 2.f * (qx * qy - qw * qz);
    float R02 = 2.f * (qx * qz + qw * qy);
    float R10 = 2.f * (qx * qy + qw * qz);
    float R11 = 1.f - 2.f * (qx * qx + qz * qz);
    float R12 = 2.f * (qy * qz - qw * qx);
    float R20 = 2.f * (qx * qz - qw * qy);
    float R21 = 2.f * (qy * qz + qw * qx);
    float R22 = 1.f - 2.f * (qx * qx + qy * qy);

    // M = R * diag(scale)
    float M00 = R00 * sx, M01 = R01 * sy, M02 = R02 * sz;
    float M10 = R10 * sx, M11 = R11 * sy, M12 = R12 * sz;
    float M20 = R20 * sx, M21 = R21 * sy, M22 = R22 * sz;

    // cov3d = M * M^T (symmetric)
    float C00 = M00 * M00 + M01 * M01 + M02 * M02;
    float C01 = M00 * M10 + M01 * M11 + M02 * M12;
    float C02 = M00 * M20 + M01 * M21 + M02 * M22;
    float C11 = M10 * M10 + M11 * M11 + M12 * M12;
    float C12 = M10 * M20 + M11 * M21 + M12 * M22;
    float C22 = M20 * M20 + M21 * M21 + M22 * M22;

    // view transform (camera 0, row-major 4x4)
    float W00 = viewmats[0],  W01 = viewmats[1],  W02 = viewmats[2],  T0 = viewmats[3];
    float W10 = viewmats[4],  W11 = viewmats[5],  W12 = viewmats[6],  T1 = viewmats[7];
    float W20 = viewmats[8],  W21 = viewmats[9],  W22 = viewmats[10], T2 = viewmats[11];

    float mx = means[n * 3 + 0], my = means[n * 3 + 1], mz = means[n * 3 + 2];
    float xc = W00 * mx + W01 * my + W02 * mz + T0;
    float yc = W10 * mx + W11 * my + W12 * mz + T1;
    float zc = W20 * mx + W21 * my + W22 * mz + T2;

    // Vc = W * C * W^T ; B = W*C first (C symmetric)
    float B00 = W00 * C00 + W01 * C01 + W02 * C02;
    float B01 = W00 * C01 + W01 * C11 + W02 * C12;
    float B02 = W00 * C02 + W01 * C12 + W02 * C22;
    float B10 = W10 * C00 + W11 * C01 + W12 * C02;
    float B11 = W10 * C01 + W11 * C11 + W12 * C12;
    float B12 = W10 * C02 + W11 * C12 + W12 * C22;
    float B20 = W20 * C00 + W21 * C01 + W22 * C02;
    float B21 = W20 * C01 + W21 * C11 + W22 * C12;
    float B22 = W20 * C02 + W21 * C12 + W22 * C22;

    float V00 = B00 * W00 + B01 * W01 + B02 * W02;
    float V01 = B00 * W10 + B01 * W11 + B02 * W12;
    float V02 = B00 * W20 + B01 * W21 + B02 * W22;
    float V11 = B10 * W10 + B11 * W11 + B12 * W12;
    float V12 = B10 * W20 + B11 * W21 + B12 * W22;
    float V22 = B20 * W20 + B21 * W21 + B22 * W22;

    float fx = Ks[0], cx = Ks[2], fy = Ks[4], cy = Ks[5];

    bool valid = zc > NEAR_Z;
    float zs = valid ? zc : 1.0f;
    float rz = 1.0f / zs;

    float u = fx * xc * rz + cx;
    float v = fy * yc * rz + cy;

    float J00 = fx * rz;
    float J02 = -fx * xc * rz * rz;
    float J11 = fy * rz;
    float J12 = -fy * yc * rz * rz;

    float a2 = J00 * J00 * V00 + 2.f * J00 * J02 * V02 + J02 * J02 * V22 + EPS2D;
    float b2 = J00 * (J11 * V01 + J12 * V02) + J02 * (J11 * V12 + J12 * V22);
    float cc2 = J11 * J11 * V11 + 2.f * J11 * J12 * V12 + J12 * J12 * V22 + EPS2D;

    float det = a2 * cc2 - b2 * b2;
    valid = valid && (det > 0.0f);
    float det_s = valid ? det : 1.0f;
    float conA = cc2 / det_s;
    float conB = -b2 / det_s;
    float conC = a2 / det_s;

    float f0 = 0.5f * conA * u * u + 0.5f * conC * v * v + conB * u * v - __logf(opac);
    float f1 = -(conA * u + conB * v);
    float f2 = 0.5f * conA;
    float f3 = -(conC * v + conB * u);
    float f4 = 0.5f * conC;
    float f5 = conB;

    if (!valid) { f0 = BIG_KEY; f1 = 0.f; f2 = 0.f; f3 = 0.f; f4 = 0.f; f5 = 0.f; }

    float* fp = feat + (size_t)n * 8;
    fp[0] = f0; fp[1] = f1; fp[2] = f2; fp[3] = f3;
    fp[4] = f4; fp[5] = f5; fp[6] = 0.f; fp[7] = 0.f;

    float* cp = col4 + (size_t)n * 4;
    cp[0] = c0; cp[1] = c1; cp[2] = c2; cp[3] = 0.f;

    key[n] = valid ? zc : BIG_KEY;
}

// ---------------------------------------------------------------------------
// Kernel 2: single-block bitonic sort of (key, index) in LDS
// ---------------------------------------------------------------------------
__global__ void gs_sort(const float* __restrict__ key, int N,
                        int* __restrict__ sortedIdx)
{
    __shared__ float skey[SORT_CAP];
    __shared__ int   sidx[SORT_CAP];
    int tid = threadIdx.x;
    for (int i = tid; i < SORT_CAP; i += blockDim.x) {
        skey[i] = (i < N) ? key[i] : BIG_KEY * 2.0f;
        sidx[i] = i;
    }
    __syncthreads();
    for (int k = 2; k <= SORT_CAP; k <<= 1) {
        for (int j = k >> 1; j > 0; j >>= 1) {
            for (int i = tid; i < SORT_CAP; i += blockDim.x) {
                int ixj = i ^ j;
                if (ixj > i) {
                    bool asc = ((i & k) == 0);
                    float ki = skey[i], kj = skey[ixj];
                    if ((ki > kj) == asc) {
                        skey[i] = kj; skey[ixj] = ki;
                        int t = sidx[i]; sidx[i] = sidx[ixj]; sidx[ixj] = t;
                    }
                }
            }
            __syncthreads();
        }
    }
    for (int i = tid; i < N; i += blockDim.x) sortedIdx[i] = sidx[i];
}

// ---------------------------------------------------------------------------
// Kernel 3: gather features/colors into sorted order, pad to tile multiple
// ---------------------------------------------------------------------------
__global__ void gs_gather(const float* __restrict__ featU,
                          const float* __restrict__ colU,
                          const int* __restrict__ sortedIdx,
                          int N, int Np,
                          float* __restrict__ featS,
                          float* __restrict__ colS)
{
    int i = blockIdx.x * blockDim.x + threadIdx.x;
    if (i >= Np) return;
    if (i < N) {
        int j = sortedIdx[i];
        #pragma unroll
        for (int k = 0; k < 8; ++k) featS[(size_t)i * 8 + k] = featU[(size_t)j * 8 + k];
        #pragma unroll
        for (int k = 0; k < 4; ++k) colS[(size_t)i * 4 + k] = colU[(size_t)j * 4 + k];
    } else {
        featS[(size_t)i * 8 + 0] = BIG_KEY;
        #pragma unroll
        for (int k = 1; k < 8; ++k) featS[(size_t)i * 8 + k] = 0.f;
        #pragma unroll
        for (int k = 0; k < 4; ++k) colS[(size_t)i * 4 + k] = 0.f;
    }
}

// ---------------------------------------------------------------------------
// Async global->LDS staging of one 16-gaussian tile (wave-uniform caller;
// EXEC all ones). Features: 32 lanes x b128 = 512B. Colors: 32 lanes x b64
// = 256B. Tracked with ASYNCcnt (CDNA5 async DMA path).
// ---------------------------------------------------------------------------
__device__ __forceinline__ void issue_tile_async(const float* featS,
                                                 const float* colS,
                                                 int t, float* dstLds, int lane)
{
    // low 32 bits of a flat LDS pointer == byte offset within group LDS
    unsigned ldsBase = (unsigned)(uintptr_t)dstLds;

    unsigned long long gf =
        (unsigned long long)(uintptr_t)(featS + (size_t)t * 128) +
        (unsigned)(lane * 16);
    unsigned lf = ldsBase + (unsigned)(lane * 16);
    asm volatile("global_load_async_to_lds_b128 %0, %1, off"
                 :: "v"(lf), "v"(gf) : "memory");

    unsigned long long gc =
        (unsigned long long)(uintptr_t)(colS + (size_t)t * 64) +
        (unsigned)(lane * 8);
    unsigned lc = ldsBase + 512u + (unsigned)(lane * 8);
    asm volatile("global_load_async_to_lds_b64 %0, %1, off"
                 :: "v"(lc), "v"(gc) : "memory");
}

__device__ __forceinline__ void wait_async_zero()
{
    asm volatile("s_wait_asynccnt 0x0" ::: "memory");
}

// ---------------------------------------------------------------------------
// Kernel 4: WMMA render. One wave = 16 pixels of a row; one block = 8 waves.
// All waves of a block share the same sorted gaussian tiles, staged into a
// double-buffered LDS tile by wave 0 via async DMA. Per tile: two
// v_wmma_f32_16x16x4_f32 -> sigma[g,p] with lane l holding pixel (l&15) and
// gaussians vbase..vbase+7 in the 8 D-VGPRs (vbase=8 for lanes 16..31).
// In-register front-to-back scan over 8 gaussians, shfl_xor(16) stitches the
// two halves in order; block-uniform early-out via __syncthreads_and.
// ---------------------------------------------------------------------------
__global__ void gs_render(const float* __restrict__ feat,   // [Np*8] sorted
                          const float* __restrict__ col4,   // [Np*4] sorted
                          const int* __restrict__ widthp,
                          int npix, int ntiles,
                          float* __restrict__ out)          // [npix*3]
{
    __shared__ float sbuf[2][192];   // [128 feat | 64 col] per buffer

    int lane = threadIdx.x & 31;
    int wid  = threadIdx.x >> 5;                 // wave in block (0..7)
    int p0   = (blockIdx.x * 8 + wid) * 16;
    bool active = (p0 < npix);                   // wave-uniform

    int width = widthp[0];
    int n  = lane & 15;   // pixel slot (WMMA N)
    int kh = lane >> 4;   // K-half selector for A/B operand layout
    int p  = p0 + n;
    int pc = (p < npix) ? p : (npix - 1);
    float px = (float)(pc % width) + 0.5f;
    float py = (float)(pc / width) + 0.5f;

    // pixel features (WMMA B: 4x16 then 4x16)
    float P[8] = {1.0f, px, px * px, py, py * py, px * py, 0.0f, 0.0f};
    v2f b0, b1;
    b0.x = P[2 * kh + 0]; b0.y = P[2 * kh + 1];
    b1.x = P[4 + 2 * kh + 0]; b1.y = P[4 + 2 * kh + 1];

    int vbase = (lane >= 16) ? 8 : 0;
    bool lowHalf = (lane < 16);

    float trun = active ? 1.0f : 0.0f;
    float accR = 0.0f, accG = 0.0f, accB = 0.0f;

    // prologue: stage tile 0
    if (wid == 0) {
        issue_tile_async(feat, col4, 0, &sbuf[0][0], lane);
        wait_async_zero();
    }
    __syncthreads();

    for (int t = 0; t < ntiles; ++t) {
        // overlap: DMA tile t+1 into the other buffer while computing tile t
        if (wid == 0 && (t + 1) < ntiles) {
            issue_tile_async(feat, col4, t + 1, &sbuf[(t + 1) & 1][0], lane);
        }

        const float* tilef = &sbuf[t & 1][0];
        const float* tilec = &sbuf[t & 1][128];

        // gaussian features (WMMA A rows: M = gaussian slot = lane&15)
        const float* fp = tilef + n * 8;
        v2f a0, a1;
        a0.x = fp[2 * kh + 0]; a0.y = fp[2 * kh + 1];
        a1.x = fp[4 + 2 * kh + 0]; a1.y = fp[4 + 2 * kh + 1];

        v8f sig = {0.f, 0.f, 0.f, 0.f, 0.f, 0.f, 0.f, 0.f};
        sig = __builtin_amdgcn_wmma_f32_16x16x4_f32(
            false, a0, false, b0, (short)0, sig, false, false);
        sig = __builtin_amdgcn_wmma_f32_16x16x4_f32(
            false, a1, false, b1, (short)0, sig, false, false);

        // local front-to-back over this half-tile's 8 gaussians (in order)
        float tl = 1.0f, cr = 0.0f, cg = 0.0f, cb = 0.0f;
        #pragma unroll
        for (int r = 0; r < 8; ++r) {
            float alpha = __expf(-sig[r]);       // opac folded into sigma
            alpha = fminf(alpha, ALPHA_MAX);
            alpha = (alpha >= ALPHA_MIN) ? alpha : 0.0f;
            const float* cp = tilec + (vbase + r) * 4;
            float w = tl * alpha;
            cr += w * cp[0];
            cg += w * cp[1];
            cb += w * cp[2];
            tl *= (1.0f - alpha);
        }

        // stitch halves: lanes 0-15 hold gaussians 0-7, lanes 16-31 hold 8-15
        float to  = __shfl_xor(tl, 16, 32);
        float orr = __shfl_xor(cr, 16, 32);
        float og  = __shfl_xor(cg, 16, 32);
        float ob  = __shfl_xor(cb, 16, 32);

        float firstT = lowHalf ? tl : to;
        float fR = lowHalf ? cr : orr;
        float fG = lowHalf ? cg : og;
        float fB = lowHalf ? cb : ob;
        float sR = lowHalf ? orr : cr;
        float sG = lowHalf ? og : cg;
        float sB = lowHalf ? ob : cb;

        float tileR = fR + firstT * sR;
        float tileG = fG + firstT * sG;
        float tileB = fB + firstT * sB;
        float tileT = tl * to;

        accR += trun * tileR;
        accG += trun * tileG;
        accB += trun * tileB;
        trun *= tileT;

        // publish: DMA for t+1 must have landed before anyone reads it, and
        // all reads of buffer (t&1) must be done before it is reused at t+2.
        if (wid == 0) wait_async_zero();
        int done = __syncthreads_and((int)(trun < 1e-6f));
        if (done) break;   // block-uniform early out
    }

    if (active && lowHalf && p < npix) {
        out[(size_t)p * 3 + 0] = accR;
        out[(size_t)p * 3 + 1] = accG;
        out[(size_t)p * 3 + 2] = accB;
    }
}

// ---------------------------------------------------------------------------
extern "C" void kernel_launch(void* const* d_in, const int* in_sizes, int n_in,
                              void* d_out, int out_size, void* d_ws, size_t ws_size,
                              hipStream_t stream)
{
    const float* means     = (const float*)d_in[0];
    const float* quats     = (const float*)d_in[1];
    const float* scales    = (const float*)d_in[2];
    const float* opacities = (const float*)d_in[3];
    const float* sh        = (const float*)d_in[4];
    const float* viewmats  = (const float*)d_in[5];
    const float* Ks        = (const float*)d_in[6];
    const int*   widthp    = (const int*)d_in[7];
    // d_in[8] = height (implied by npix/width)

    int N    = in_sizes[3];                 // number of gaussians
    int Np   = ((N + 15) / 16) * 16;        // padded to 16-gaussian tiles
    int npix = out_size / 3;

    float* ws    = (float*)d_ws;
    float* featU = ws;             ws += (size_t)Np * 8;
    float* colU  = ws;             ws += (size_t)Np * 4;
    float* key   = ws;             ws += (size_t)Np;
    int*   sidx  = (int*)ws;       ws += (size_t)Np;
    float* featS = ws;             ws += (size_t)Np * 8;
    float* colS  = ws;             ws += (size_t)Np * 4;

    gs_preprocess<<<(N + 255) / 256, 256, 0, stream>>>(
        means, quats, scales, opacities, sh, viewmats, Ks, N, featU, colU, key);

    gs_sort<<<1, 256, 0, stream>>>(key, N, sidx);

    gs_gather<<<(Np + 255) / 256, 256, 0, stream>>>(
        featU, colU, sidx, N, Np, featS, colS);

    int waves  = (npix + 15) / 16;
    int blocks = (waves + 7) / 8;           // 256 threads = 8 waves per block
    gs_render<<<blocks, 256, 0, stream>>>(
        featS, colS, widthp, npix, Np / 16, (float*)d_out);
}